// GraphSAGE_23673859735793
// MI455X (gfx1250) — compile-verified
//
#include <hip/hip_runtime.h>
#include <hip/hip_bf16.h>

typedef float v2f __attribute__((ext_vector_type(2)));
typedef float v8f __attribute__((ext_vector_type(8)));

#define IN_CH 128   // K dimension for both layers (HID_CH == IN_CH == 128)

// ---------------------------------------------------------------------------
// Zero-fill workspace (agg buffers + degree) — required every call since the
// scatter kernels accumulate with atomics.
// ---------------------------------------------------------------------------
__global__ void sage_zero_f32(float* __restrict__ p, long long n) {
    long long i = (long long)blockIdx.x * blockDim.x + threadIdx.x;
    if (i < n) p[i] = 0.0f;
}

// ---------------------------------------------------------------------------
// Edge scatter: agg[dst, :] += feat[src, :]; optionally deg[dst] += 1.
// One thread handles one (edge, 4-channel group): float4 gather from the
// source row, four global_atomic_add_f32 into the destination row.
// 800k * 32 threads.
// ---------------------------------------------------------------------------
__global__ void sage_scatter_add(const float* __restrict__ feat,   // [N,128]
                                 const int*   __restrict__ src,
                                 const int*   __restrict__ dst,
                                 float*       __restrict__ agg,    // [N,128]
                                 float*       __restrict__ deg,    // [N] or null
                                 int n_edges) {
    int t = blockIdx.x * blockDim.x + threadIdx.x;
    int e = t >> 5;
    if (e >= n_edges) return;
    int g = t & 31;                       // which group of 4 channels
    int s = src[e];
    int d = dst[e];
    const float4* frow = (const float4*)(feat + (size_t)s * IN_CH);
    float4 v = frow[g];
    float* arow = agg + (size_t)d * IN_CH + g * 4;
    atomicAdd(arow + 0, v.x);
    atomicAdd(arow + 1, v.y);
    atomicAdd(arow + 2, v.z);
    atomicAdd(arow + 3, v.w);
    if (deg != nullptr && g == 0) atomicAdd(deg + d, 1.0f);
}

// ---------------------------------------------------------------------------
// Fused SAGE layer via V_WMMA_F32_16X16X4_F32 (full f32 precision):
//   out = act( (agg * 1/max(deg,1)) @ Wl + bias + xroot @ Wr )
// One wave computes a 16x16 output tile; block = NCOLS/16 waves covers
// 16 rows x NCOLS. K = 128, unrolled as 2 x 32 WMMA issues per wave.
//
// Fragment layouts per CDNA5 ISA 7.12.2 (wave32):
//   A (16x4, MxK): lane = (k>=2 ? 16:0) + m ; vgpr0=K even, vgpr1=K odd
//   B (4x16, KxN): lane = (k>=2 ? 16:0) + n ; vgpr r = row khalf*2+r
//   C/D (16x16):   lane = (m>=8 ? 16:0) + n ; vgpr v = row v + khalf*8
// Requires gridDim.x * 16 == n_nodes (50000 = 16 * 3125) so EXEC is all 1s.
// ---------------------------------------------------------------------------
template <int NCOLS, bool RELU>
__global__ void sage_layer_wmma(const float* __restrict__ agg,   // [N,128]
                                const float* __restrict__ deg,   // [N]
                                const float* __restrict__ xroot, // [N,128]
                                const float* __restrict__ Wl,    // [128,NCOLS]
                                const float* __restrict__ bias,  // [NCOLS]
                                const float* __restrict__ Wr,    // [128,NCOLS]
                                float*       __restrict__ out) { // [N,NCOLS]
    const int wave  = threadIdx.x >> 5;       // N-tile within the row block
    const int lane  = threadIdx.x & 31;
    const int m0    = blockIdx.x * 16;
    const int n0    = wave * 16;
    const int mrow  = m0 + (lane & 15);       // A row for this lane
    const int khalf = lane >> 4;              // 0: K{0,1}, 1: K{2,3}
    const int ncol  = n0 + (lane & 15);       // B/C/D column for this lane

    const float rdeg = 1.0f / fmaxf(deg[mrow], 1.0f);

    const float* aggRow = agg   + (size_t)mrow * IN_CH;
    const float* xRow   = xroot + (size_t)mrow * IN_CH;

    v8f acc = {};

    // mean @ Wl  (mean computed on the fly: agg * rdeg)
    #pragma unroll
    for (int k0 = 0; k0 < IN_CH; k0 += 4) {
        const int ka = k0 + khalf * 2;
        v2f a, b;
        a.x = aggRow[ka + 0] * rdeg;
        a.y = aggRow[ka + 1] * rdeg;
        b.x = Wl[(size_t)(ka + 0) * NCOLS + ncol];
        b.y = Wl[(size_t)(ka + 1) * NCOLS + ncol];
        acc = __builtin_amdgcn_wmma_f32_16x16x4_f32(
            /*neg_a=*/false, a, /*neg_b=*/false, b,
            /*c_mod=*/(short)0, acc, /*reuse_a=*/false, /*reuse_b=*/false);
    }
    // x @ Wr, accumulating into the same tile
    #pragma unroll
    for (int k0 = 0; k0 < IN_CH; k0 += 4) {
        const int ka = k0 + khalf * 2;
        v2f a, b;
        a.x = xRow[ka + 0];
        a.y = xRow[ka + 1];
        b.x = Wr[(size_t)(ka + 0) * NCOLS + ncol];
        b.y = Wr[(size_t)(ka + 1) * NCOLS + ncol];
        acc = __builtin_amdgcn_wmma_f32_16x16x4_f32(
            false, a, false, b, (short)0, acc, false, false);
    }

    const float bv = bias[ncol];
    #pragma unroll
    for (int v = 0; v < 8; ++v) {
        const int row = m0 + v + khalf * 8;
        float r = acc[v] + bv;
        if (RELU) r = fmaxf(r, 0.0f);
        out[(size_t)row * NCOLS + ncol] = r;
    }
}

// ---------------------------------------------------------------------------
// Inputs (setup_inputs order): x, edge_index, W1_l, b1, W1_r, W2_l, b2, W2_r
// Output: [N, 64] float32.
// ---------------------------------------------------------------------------
extern "C" void kernel_launch(void* const* d_in, const int* in_sizes, int n_in,
                              void* d_out, int out_size, void* d_ws, size_t ws_size,
                              hipStream_t stream) {
    const float* x    = (const float*)d_in[0];
    const int*   ei   = (const int*)  d_in[1];   // [2, E] row-major
    const float* W1_l = (const float*)d_in[2];
    const float* b1   = (const float*)d_in[3];
    const float* W1_r = (const float*)d_in[4];
    const float* W2_l = (const float*)d_in[5];
    const float* b2   = (const float*)d_in[6];
    const float* W2_r = (const float*)d_in[7];
    float* out = (float*)d_out;

    const int N  = in_sizes[0] / IN_CH;   // 50000
    const int NE = in_sizes[1] / 2;       // 800000
    const int* src = ei;
    const int* dst = ei + NE;

    // Workspace layout (floats): deg[N] | agg1[N*128] | h[N*128] | agg2[N*128]
    float* deg  = (float*)d_ws;
    float* agg1 = deg  + N;
    float* h    = agg1 + (size_t)N * IN_CH;
    float* agg2 = h    + (size_t)N * IN_CH;

    // 1) zero deg/agg1/h/agg2 (contiguous region)
    {
        long long nz = (long long)N * (1 + 3 * IN_CH);
        int blocks = (int)((nz + 255) / 256);
        sage_zero_f32<<<blocks, 256, 0, stream>>>((float*)d_ws, nz);
    }
    // 2) layer-1 scatter: agg1 += x[src] per dst; deg counted once per edge
    {
        long long nt = (long long)NE * 32;
        int blocks = (int)((nt + 255) / 256);
        sage_scatter_add<<<blocks, 256, 0, stream>>>(x, src, dst, agg1, deg, NE);
    }
    // 3) layer 1: h = relu(mean1 @ W1_l + b1 + x @ W1_r)   [N,128]
    sage_layer_wmma<128, true><<<N / 16, 32 * (128 / 16), 0, stream>>>(
        agg1, deg, x, W1_l, b1, W1_r, h);
    // 4) layer-2 scatter: agg2 += h[src] per dst
    {
        long long nt = (long long)NE * 32;
        int blocks = (int)((nt + 255) / 256);
        sage_scatter_add<<<blocks, 256, 0, stream>>>(h, src, dst, agg2, nullptr, NE);
    }
    // 5) layer 2: out = mean2 @ W2_l + b2 + h @ W2_r       [N,64]
    sage_layer_wmma<64, false><<<N / 16, 32 * (64 / 16), 0, stream>>>(
        agg2, deg, h, W2_l, b2, W2_r, out);
}